// AttentiveTransformer_43568148251020
// MI455X (gfx1250) — compile-verified
//
#include <hip/hip_runtime.h>
#include <hip/hip_bf16.h>
#include <math.h>
#include <stdint.h>

typedef __attribute__((ext_vector_type(2))) float v2f;
typedef __attribute__((ext_vector_type(8))) float v8f;
typedef int v4i_g __attribute__((__vector_size__(4 * sizeof(int))));   // builtin param elem type
typedef __attribute__((address_space(3))) v4i_g v4i_lds;               // same, in LDS (AS3)

#define BATCH   65536
#define KDIM    512
#define GCOLS   2048
#define VBS_SZ  128
#define BN_EPS  1e-5f

#define TM   128   // block tile M == VBS (enables fused ghost-BN stats)
#define TN   128   // block tile N
#define TK   16    // K step staged through LDS
#define PADF 20    // LDS row stride in floats: 80B keeps B128 16B-aligned AND
                   // r*20 mod 64 hits 16 distinct banks -> conflict-free frags
#define NKT  (KDIM / TK)

// ---- CDNA5 async global->LDS path (guarded; falls back to ld+ds_store) ----
#if __has_builtin(__builtin_amdgcn_global_load_async_to_lds_b128)
#define HAS_ASYNC_LDS 1
#else
#define HAS_ASYNC_LDS 0
#endif

__device__ __forceinline__ v4i_g* as_g16(const void* p) {
    // strip const; pointer to 16B vector (builtin param type)
    return (v4i_g*)(uintptr_t)p;
}
__device__ __forceinline__ v4i_lds* as_l16(void* p) {
    // generic LDS addr = {shared_aperture_hi, lds_offset[31:0]} -> truncate to AS3
    return (v4i_lds*)(unsigned int)(uintptr_t)p;
}

__device__ __forceinline__ void wait_stage() {
#if HAS_ASYNC_LDS
#if __has_builtin(__builtin_amdgcn_s_wait_asynccnt)
    __builtin_amdgcn_s_wait_asynccnt(0);
#else
    asm volatile("s_wait_asynccnt 0" ::: "memory");
#endif
#endif
    __syncthreads();
}

// Stage one 128x16 A tile and 128x16 B tile into LDS (async when available).
__device__ __forceinline__ void stage_tile(const float* __restrict__ gA,
                                           const float* __restrict__ gW,
                                           float* lA, float* lB,
                                           int mBase, int nBase, int k0, int tid)
{
    #pragma unroll
    for (int i = 0; i < 2; ++i) {
        const int lin = tid + i * 256;     // 0..511
        const int r   = lin >> 2;          // 0..127 row
        const int kq  = (lin & 3) << 2;    // 0,4,8,12 float offset (16B quad)
        const float* srcA = gA + (size_t)(mBase + r) * KDIM + k0 + kq;
        const float* srcB = gW + (size_t)(nBase + r) * KDIM + k0 + kq;
        float* dstA = lA + r * PADF + kq;
        float* dstB = lB + r * PADF + kq;
#if HAS_ASYNC_LDS
        __builtin_amdgcn_global_load_async_to_lds_b128(as_g16(srcA), as_l16(dstA), 0, 0);
        __builtin_amdgcn_global_load_async_to_lds_b128(as_g16(srcB), as_l16(dstB), 0, 0);
#else
        *(float4*)dstA = *(const float4*)srcA;
        *(float4*)dstB = *(const float4*)srcB;
#endif
    }
}

__global__ __launch_bounds__(256)
void at_gemm_stats(const float* __restrict__ A,     // [B, K] processed_feat
                   const float* __restrict__ W,     // [G, K]
                   float* __restrict__ X,           // [B, G] workspace
                   float* __restrict__ meanOut,     // [B/VBS, G]
                   float* __restrict__ rstdOut)     // [B/VBS, G]
{
    __shared__ float As[2][TM * PADF];
    __shared__ float Bs[2][TN * PADF];
    __shared__ float cSum[TN];
    __shared__ float cSq[TN];

    const int tid  = threadIdx.x;
    const int lane = tid & 31;
    const int wid  = tid >> 5;
    const int wm   = wid >> 2;      // 0..1 : 64 output rows per wave
    const int wn   = wid & 3;       // 0..3 : 32 output cols per wave
    const int lm   = lane & 15;
    const int lh   = lane >> 4;

    const int nBase = blockIdx.x * TN;
    const int mBase = blockIdx.y * TM;

    if (tid < TN) { cSum[tid] = 0.f; cSq[tid] = 0.f; }

    const v8f vzero = {0.f,0.f,0.f,0.f,0.f,0.f,0.f,0.f};
    v8f acc[4][2];
    #pragma unroll
    for (int mt = 0; mt < 4; ++mt)
        #pragma unroll
        for (int nt = 0; nt < 2; ++nt)
            acc[mt][nt] = vzero;

    // Prime the pipeline: tile 0 -> buffer 0
    stage_tile(A, W, As[0], Bs[0], mBase, nBase, 0, tid);

    for (int kt = 0; kt < NKT; ++kt) {
        wait_stage();                       // my async copies done + all waves synced
        const int cur = kt & 1;
        if (kt + 1 < NKT)                   // prefetch next tile into other buffer
            stage_tile(A, W, As[cur ^ 1], Bs[cur ^ 1],
                       mBase, nBase, (kt + 1) * TK, tid);

        const float* lA = As[cur];
        const float* lB = Bs[cur];

        #pragma unroll
        for (int kk = 0; kk < TK; kk += 4) {
            const int ks = kk + lh * 2;     // ISA A-16x4 layout: lane half -> K pair
            v2f af[4];
            #pragma unroll
            for (int mt = 0; mt < 4; ++mt) {
                const int m = wm * 64 + mt * 16 + lm;
                af[mt].x = lA[m * PADF + ks];
                af[mt].y = lA[m * PADF + ks + 1];
            }
            v2f bf[2];
            #pragma unroll
            for (int nt = 0; nt < 2; ++nt) {
                const int n = wn * 32 + nt * 16 + lm;
                bf[nt].x = lB[n * PADF + ks];
                bf[nt].y = lB[n * PADF + ks + 1];
            }
            #pragma unroll
            for (int mt = 0; mt < 4; ++mt)
                #pragma unroll
                for (int nt = 0; nt < 2; ++nt)
                    acc[mt][nt] = __builtin_amdgcn_wmma_f32_16x16x4_f32(
                        false, af[mt], false, bf[nt], (short)0, acc[mt][nt],
                        false, false);
        }
    }

    // Store X tile (D layout: VGPR v -> M = v + 8*laneHalf, N = lane&15)
    #pragma unroll
    for (int mt = 0; mt < 4; ++mt) {
        #pragma unroll
        for (int nt = 0; nt < 2; ++nt) {
            const int col = nBase + wn * 32 + nt * 16 + lm;
            #pragma unroll
            for (int v = 0; v < 8; ++v) {
                const int row = mBase + wm * 64 + mt * 16 + lh * 8 + v;
                X[(size_t)row * GCOLS + col] = acc[mt][nt][v];
            }
        }
    }

    // Fused ghost-batch stats: this block owns exactly one 128-row VBS chunk.
    #pragma unroll
    for (int nt = 0; nt < 2; ++nt) {
        float s = 0.f, ss = 0.f;
        #pragma unroll
        for (int mt = 0; mt < 4; ++mt)
            #pragma unroll
            for (int v = 0; v < 8; ++v) {
                const float a = acc[mt][nt][v];
                s += a; ss += a * a;
            }
        s  += __shfl_xor(s, 16, 32);    // fold lane halves (same N column)
        ss += __shfl_xor(ss, 16, 32);
        if (lane < 16) {
            atomicAdd(&cSum[wn * 32 + nt * 16 + lane], s);   // ds_add_f32 x 2 M-waves
            atomicAdd(&cSq [wn * 32 + nt * 16 + lane], ss);
        }
    }
    __syncthreads();
    if (tid < TN) {
        const float m  = cSum[tid] * (1.f / VBS_SZ);
        const float vv = fmaxf(cSq[tid] * (1.f / VBS_SZ) - m * m, 0.f);
        meanOut[(size_t)blockIdx.y * GCOLS + nBase + tid] = m;
        rstdOut[(size_t)blockIdx.y * GCOLS + nBase + tid] = rsqrtf(vv + BN_EPS);
    }
}

__global__ __launch_bounds__(256)
void at_bn_sparsemax(const float* __restrict__ X,
                     const float* __restrict__ priors,
                     const float* __restrict__ gamma,
                     const float* __restrict__ beta,
                     const float* __restrict__ meanIn,
                     const float* __restrict__ rstdIn,
                     float* __restrict__ out)
{
    __shared__ float red[8];
    const int row  = blockIdx.x;
    const int tid  = threadIdx.x;
    const int lane = tid & 31;
    const int wid  = tid >> 5;
    const size_t rbase = (size_t)row * GCOLS;
    const size_t sbase = (size_t)(row >> 7) * GCOLS;   // ghost chunk index

    float z[8];
    #pragma unroll
    for (int i = 0; i < 8; ++i) {
        const int col = tid + i * 256;
        const float x  = X[rbase + col];
        const float zn = (x - meanIn[sbase + col]) * rstdIn[sbase + col];
        z[i] = (zn * gamma[col] + beta[col]) * priors[rbase + col];
    }

    // Row max
    float mx = z[0];
    #pragma unroll
    for (int i = 1; i < 8; ++i) mx = fmaxf(mx, z[i]);
    #pragma unroll
    for (int off = 16; off >= 1; off >>= 1) mx = fmaxf(mx, __shfl_xor(mx, off, 32));
    if (lane == 0) red[wid] = mx;
    __syncthreads();
    if (tid == 0) {
        float m = red[0];
        for (int j = 1; j < 8; ++j) m = fmaxf(m, red[j]);
        red[0] = m;
    }
    __syncthreads();
    mx = red[0];

    // Sparsemax threshold: f(tau)=sum(relu(z-tau)) monotone; f(mx-1)>=1, f(mx)=0
    float lo = mx - 1.f, hi = mx;
    for (int it = 0; it < 32; ++it) {
        const float tauM = 0.5f * (lo + hi);
        float s = 0.f;
        #pragma unroll
        for (int i = 0; i < 8; ++i) s += fmaxf(z[i] - tauM, 0.f);
        #pragma unroll
        for (int off = 16; off >= 1; off >>= 1) s += __shfl_xor(s, off, 32);
        __syncthreads();                 // protect red[] reuse
        if (lane == 0) red[wid] = s;
        __syncthreads();
        if (tid == 0) {
            float t = 0.f;
            for (int j = 0; j < 8; ++j) t += red[j];
            red[0] = t;
        }
        __syncthreads();
        const float total = red[0];
        if (total >= 1.f) lo = tauM; else hi = tauM;
    }
    const float tau = 0.5f * (lo + hi);
    #pragma unroll
    for (int i = 0; i < 8; ++i) {
        const int col = tid + i * 256;
        out[rbase + col] = fmaxf(z[i] - tau, 0.f);
    }
}

extern "C" void kernel_launch(void* const* d_in, const int* in_sizes, int n_in,
                              void* d_out, int out_size, void* d_ws, size_t ws_size,
                              hipStream_t stream) {
    (void)in_sizes; (void)n_in; (void)out_size; (void)ws_size;
    const float* priors = (const float*)d_in[0];
    const float* feat   = (const float*)d_in[1];
    const float* W      = (const float*)d_in[2];
    const float* gamma  = (const float*)d_in[3];
    const float* beta   = (const float*)d_in[4];
    float* out = (float*)d_out;

    float* X    = (float*)d_ws;                                  // [B, G]
    float* mean = X    + (size_t)BATCH * GCOLS;                  // [B/VBS, G]
    float* rstd = mean + (size_t)(BATCH / VBS_SZ) * GCOLS;       // [B/VBS, G]

    dim3 g1(GCOLS / TN, BATCH / TM);
    at_gemm_stats<<<g1, 256, 0, stream>>>(feat, W, X, mean, rstd);
    at_bn_sparsemax<<<BATCH, 256, 0, stream>>>(X, priors, gamma, beta, mean, rstd, out);
}